// QuantumRegressor_88338887344763
// MI455X (gfx1250) — compile-verified
//
#include <hip/hip_runtime.h>
#include <hip/hip_bf16.h>

#define NQ     12
#define DIM    4096          // 2^12
#define NL     6
#define BLOCK  256
#define PER    (DIM / BLOCK) // 16 amplitudes per thread

typedef __attribute__((ext_vector_type(2))) float v2f;
typedef __attribute__((ext_vector_type(8))) float v8f;

__global__ __launch_bounds__(BLOCK)
void qsim_kernel(const float* __restrict__ x,        // [B,12]
                 const float* __restrict__ weights,  // [6,12,3]
                 const float* __restrict__ Wv,       // [1,12]
                 const float* __restrict__ bv,       // [1]
                 float* __restrict__ out)            // [B]
{
    __shared__ float sRe[DIM];
    __shared__ float sIm[DIM];
    __shared__ float sUr[3][16][16];   // fused 16x16 group unitaries (real)
    __shared__ float sUi[3][16][16];   // (imag)
    __shared__ float gRe[NQ][4];       // per-wire 2x2 Rot gate entries
    __shared__ float gIm[NQ][4];
    __shared__ float cw[NQ], sw[NQ];   // initial RY cos/sin
    __shared__ float wW[NQ];           // readout weights
    __shared__ float red[BLOCK];

    const int tid   = threadIdx.x;
    const int batch = blockIdx.x;
    const int lane  = tid & 31;
    const int wave  = tid >> 5;

    // ---- load per-batch angles + readout weights ----
    if (tid < NQ) {
        float h = 0.5f * x[batch * NQ + tid];
        cw[tid] = cosf(h);
        sw[tid] = sinf(h);
        wW[tid] = Wv[tid];
    }
    __syncthreads();

    // ---- initial product state: psi[i] = prod_w (bit ? s : c), imag = 0 ----
    for (int k = 0; k < PER; ++k) {
        int i = k * BLOCK + tid;
        float a = 1.0f;
        #pragma unroll
        for (int w = 0; w < NQ; ++w)
            a *= ((i >> (11 - w)) & 1) ? sw[w] : cw[w];
        sRe[i] = a;
        sIm[i] = 0.0f;
    }
    __syncthreads();

    // ---- layers ----
    for (int l = 0; l < NL; ++l) {
        // per-wire Rot gate: m00=ep*ct, m01=-em*st, m10=conj(em)*st, m11=conj(ep)*ct
        // ep = exp(-0.5i(phi+omega)), em = exp(+0.5i(phi-omega))
        if (tid < NQ) {
            const float* wl = weights + (l * NQ + tid) * 3;
            float phi = wl[0], th = wl[1], om = wl[2];
            float ct = cosf(0.5f * th), st = sinf(0.5f * th);
            float ap = 0.5f * (phi + om), am = 0.5f * (phi - om);
            float cp = cosf(ap), sp = sinf(ap);
            float cm = cosf(am), sm = sinf(am);
            gRe[tid][0] =  cp * ct;  gIm[tid][0] = -sp * ct;  // m00
            gRe[tid][1] = -cm * st;  gIm[tid][1] = -sm * st;  // m01
            gRe[tid][2] =  cm * st;  gIm[tid][2] = -sm * st;  // m10
            gRe[tid][3] =  cp * ct;  gIm[tid][3] =  sp * ct;  // m11
        }
        __syncthreads();

        // build fused 16x16 unitaries: U_g = G_{4g} (x) G_{4g+1} (x) G_{4g+2} (x) G_{4g+3}
        {
            int i = tid >> 4, j = tid & 15;
            #pragma unroll
            for (int g = 0; g < 3; ++g) {
                float ur = 1.0f, ui = 0.0f;
                #pragma unroll
                for (int kq = 0; kq < 4; ++kq) {
                    int w  = 4 * g + kq;
                    int ib = (i >> (3 - kq)) & 1;
                    int jb = (j >> (3 - kq)) & 1;
                    int e  = ib * 2 + jb;
                    float gr = gRe[w][e], gi = gIm[w][e];
                    float nr = ur * gr - ui * gi;
                    float ni = ur * gi + ui * gr;
                    ur = nr; ui = ni;
                }
                sUr[g][i][j] = ur;
                sUi[g][i][j] = ui;
            }
        }
        __syncthreads();

        // apply the three 16x16 complex gates via V_WMMA_F32_16X16X4_F32
        // psi reshaped [pre,16,post]; columns c = pre*post + post_idx, 256 columns,
        // 16 tiles of 16 columns; wave handles tiles {2*wave, 2*wave+1} (disjoint cols)
        #pragma unroll
        for (int g = 0; g < 3; ++g) {
            const int shift = 8 - 4 * g;          // log2(post size)
            const int pmask = (1 << shift) - 1;
            const int m     = lane & 15;          // A-row / D-column-lane id
            const int khalf = (lane >> 4) << 1;   // 0 or 2 (K pair per lane half)
            const int mo    = (lane >> 4) << 3;   // D: +8 rows for upper lanes
            #pragma unroll
            for (int tt = 0; tt < 2; ++tt) {
                int tile = wave * 2 + tt;
                int col  = tile * 16 + m;
                int base = ((col >> shift) << (shift + 4)) + (col & pmask);
                v8f dr = {0.f,0.f,0.f,0.f,0.f,0.f,0.f,0.f};
                v8f di = {0.f,0.f,0.f,0.f,0.f,0.f,0.f,0.f};
                #pragma unroll
                for (int kk = 0; kk < 16; kk += 4) {
                    int k0 = kk + khalf, k1 = k0 + 1;
                    v2f ar  = {  sUr[g][m][k0],  sUr[g][m][k1] };
                    v2f ai  = {  sUi[g][m][k0],  sUi[g][m][k1] };
                    v2f ain = { -sUi[g][m][k0], -sUi[g][m][k1] };
                    int f0 = base + (k0 << shift);
                    int f1 = base + (k1 << shift);
                    v2f br = { sRe[f0], sRe[f1] };
                    v2f bi = { sIm[f0], sIm[f1] };
                    // Dr += Ur*Xr - Ui*Xi ; Di += Ur*Xi + Ui*Xr
                    dr = __builtin_amdgcn_wmma_f32_16x16x4_f32(false, ar,  false, br, (short)0, dr, false, false);
                    dr = __builtin_amdgcn_wmma_f32_16x16x4_f32(false, ain, false, bi, (short)0, dr, false, false);
                    di = __builtin_amdgcn_wmma_f32_16x16x4_f32(false, ar,  false, bi, (short)0, di, false, false);
                    di = __builtin_amdgcn_wmma_f32_16x16x4_f32(false, ai,  false, br, (short)0, di, false, false);
                }
                // scatter D back: VGPR v -> row m = v + mo, same column
                #pragma unroll
                for (int v = 0; v < 8; ++v) {
                    int f = base + ((v + mo) << shift);
                    sRe[f] = dr[v];
                    sIm[f] = di[v];
                }
            }
            __syncthreads();
        }

        // CNOT ring: control w, target (w+r)%12, as in-place swaps
        int r = l % (NQ - 1) + 1;
        for (int w = 0; w < NQ; ++w) {
            int cmask = 1 << (11 - w);
            int tmask = 1 << (11 - ((w + r) % NQ));
            #pragma unroll
            for (int k = 0; k < PER; ++k) {
                int i = k * BLOCK + tid;
                if ((i & cmask) && !(i & tmask)) {
                    int j = i | tmask;
                    float tr = sRe[i], ti = sIm[i];
                    sRe[i] = sRe[j]; sIm[i] = sIm[j];
                    sRe[j] = tr;     sIm[j] = ti;
                }
            }
            __syncthreads();
        }
    }

    // ---- readout: y = b + sum_i |psi_i|^2 * sum_w W_w*(1-2*bit_w(i)) ----
    float acc = 0.0f;
    for (int k = 0; k < PER; ++k) {
        int i = k * BLOCK + tid;
        float p = sRe[i] * sRe[i] + sIm[i] * sIm[i];
        float coef = 0.0f;
        #pragma unroll
        for (int w = 0; w < NQ; ++w)
            coef += ((i >> (11 - w)) & 1) ? -wW[w] : wW[w];
        acc += p * coef;
    }
    red[tid] = acc;
    __syncthreads();
    for (int s2 = BLOCK / 2; s2 > 0; s2 >>= 1) {
        if (tid < s2) red[tid] += red[tid + s2];
        __syncthreads();
    }
    if (tid == 0) out[batch] = red[0] + bv[0];
}

extern "C" void kernel_launch(void* const* d_in, const int* in_sizes, int n_in,
                              void* d_out, int out_size, void* d_ws, size_t ws_size,
                              hipStream_t stream) {
    const float* x       = (const float*)d_in[0];  // [B,12]
    const float* weights = (const float*)d_in[1];  // [6,12,3]
    const float* W       = (const float*)d_in[2];  // [1,12]
    const float* b       = (const float*)d_in[3];  // [1]
    float* out = (float*)d_out;                    // [B]
    int batches = in_sizes[0] / NQ;
    qsim_kernel<<<batches, BLOCK, 0, stream>>>(x, weights, W, b, out);
    (void)n_in; (void)out_size; (void)d_ws; (void)ws_size;
}